// KENN_GCN_18992345383146
// MI455X (gfx1250) — compile-verified
//
#include <hip/hip_runtime.h>
#include <hip/hip_bf16.h>
#include <math.h>
#include <stdint.h>

// Problem constants (match reference)
#define NN 50000
#define EE 1600000
#define HH 128
#define CC 40
#define NUM_KENN 3
#define EPSB 1e-5f

typedef __attribute__((ext_vector_type(2))) float v2f;
typedef __attribute__((ext_vector_type(8))) float v8f;

// ---------------------------------------------------------------------------
// Async global->LDS staging (CDNA5 GLOBAL_LOAD_ASYNC_TO_LDS_B128, ASYNCcnt)
// Builtin signature (from hipcc diagnostic): param0 is
//   'int __attribute__((vector_size(16))) __device__ *'  (v4i32, addrspace(1))
// so param1 is the matching v4i32 addrspace(3) LDS pointer.
// Note: per ISA 10.2, flat-address low 32 bits == LDS byte offset, so the
// uintptr_t-based addrspace conversion is well-defined here.
// ---------------------------------------------------------------------------
#if __has_builtin(__builtin_amdgcn_global_load_async_to_lds_b128)
#define HAS_ASYNC_LDS 1
#else
#define HAS_ASYNC_LDS 0
#endif

typedef int vi4 __attribute__((vector_size(16)));
typedef __attribute__((address_space(1))) vi4 gvi4;   // global v4i32
typedef __attribute__((address_space(3))) vi4 lvi4;   // LDS v4i32

__device__ __forceinline__ void cp16_async(float* l, const float* g) {
#if HAS_ASYNC_LDS
    __builtin_amdgcn_global_load_async_to_lds_b128(
        (gvi4*)(uintptr_t)g, (lvi4*)(uintptr_t)l, 0, 0);
#else
    *(float4*)l = *(const float4*)g;
#endif
}

__device__ __forceinline__ void async_stage_wait() {
#if HAS_ASYNC_LDS
#if __has_builtin(__builtin_amdgcn_s_wait_asynccnt)
    __builtin_amdgcn_s_wait_asynccnt(0);
#else
    asm volatile("s_wait_asynccnt 0" ::: "memory");
#endif
#endif
}

// ---------------------------------------------------------------------------
// WMMA GEMM: out[N x NC] = A[N x 128] @ W[128 x NC]
// Block = NTILES waves; each block covers MSUB*16 = 80 rows (NN/80 = 625
// blocks exactly). W tile staged zero-padded in LDS (no divergent loads in
// the K loop), A tile staged with padded stride 132 (bank-conflict free).
// 5 independent accumulator chains interleaved per K-step for XDL ILP.
// ---------------------------------------------------------------------------
#define MSUB 5
#define MROWS (MSUB * 16)

template <int NC, int NTILES>
__global__ __launch_bounds__(NTILES * 32) void gemm_wmma(
    const float* __restrict__ A, const float* __restrict__ W,
    float* __restrict__ out) {
    constexpr int WC = NTILES * 16;        // padded W width
    __shared__ float As[MROWS * 132];
    __shared__ float Ws[128 * WC];

    const int m0 = blockIdx.x * MROWS;

    // Stage A tile: MROWS x 128 floats as 16B async copies.
    for (int v4 = threadIdx.x; v4 < (MROWS * 128) / 4; v4 += blockDim.x) {
        int v = v4 << 2;
        int r = v >> 7, c = v & 127;
        cp16_async(&As[r * 132 + c], &A[(size_t)(m0 + r) * 128 + c]);
    }
    // Stage W tile (zero-padded to WC columns when NC < WC).
    if (NC == WC) {
        for (int v4 = threadIdx.x; v4 < (128 * WC) / 4; v4 += blockDim.x)
            cp16_async(&Ws[v4 << 2], &W[v4 << 2]);
    } else {
        for (int idx = threadIdx.x; idx < 128 * WC; idx += blockDim.x) {
            int r = idx / WC, c = idx - r * WC;
            Ws[idx] = (c < NC) ? W[(size_t)r * NC + c] : 0.f;
        }
    }
    async_stage_wait();
    __syncthreads();

    const int lane = threadIdx.x & 31;
    const int wave = threadIdx.x >> 5;      // == n-tile
    const int half = lane >> 4;             // 0: lanes 0-15, 1: lanes 16-31
    const int l16  = lane & 15;
    const int col  = wave * 16 + l16;       // column in padded Ws

    v8f acc[MSUB];
#pragma unroll
    for (int m = 0; m < MSUB; ++m) acc[m] = (v8f){};

    for (int k = 0; k < 128; k += 4) {
        const int ka = k + (half << 1);     // K pair handled by this half-wave
        v2f b;
        b.x = Ws[ka * WC + col];
        b.y = Ws[(ka + 1) * WC + col];
#pragma unroll
        for (int m = 0; m < MSUB; ++m) {
            const float* ap = &As[(m * 16 + l16) * 132 + ka];
            v2f a;
            a.x = ap[0];
            a.y = ap[1];
            acc[m] = __builtin_amdgcn_wmma_f32_16x16x4_f32(
                false, a, false, b, (short)0, acc[m], false, false);
        }
    }

    if (col < NC) {
#pragma unroll
        for (int m = 0; m < MSUB; ++m)
#pragma unroll
            for (int i = 0; i < 8; ++i)  // D: VGPR i -> row i / 8+i per half
                out[(size_t)(m0 + m * 16 + (half << 3) + i) * NC + col] =
                    acc[m][i];
    }
}

// ---------------------------------------------------------------------------
// Small utility kernels
// ---------------------------------------------------------------------------
__global__ void fill_kernel(float* __restrict__ p, float v, long n) {
    long i = blockIdx.x * (long)blockDim.x + threadIdx.x;
    if (i < n) p[i] = v;
}

__global__ void copy_kernel(float* __restrict__ d, const float* __restrict__ s, long n) {
    long i = blockIdx.x * (long)blockDim.x + threadIdx.x;
    if (i < n) d[i] = s[i];
}

__global__ void deg_kernel(float* __restrict__ deg, const int* __restrict__ dst) {
    int e = blockIdx.x * blockDim.x + threadIdx.x;
    if (e < EE) atomicAdd(&deg[dst[e]], 1.f);
}

__global__ void rsqrt_kernel(float* __restrict__ p, int n) {
    int i = blockIdx.x * blockDim.x + threadIdx.x;
    if (i < n) p[i] = rsqrtf(p[i]);
}

// agg = msg * dinv^2 + bias   (self-loop term + bias)
template <int NC>
__global__ void agg_init(const float* __restrict__ msg, float* __restrict__ agg,
                         const float* __restrict__ dinv, const float* __restrict__ bias,
                         long total) {
    long idx = blockIdx.x * (long)blockDim.x + threadIdx.x;
    if (idx >= total) return;
    long i = idx / NC;
    int  c = (int)(idx - i * NC);
    float di = dinv[i];
    agg[idx] = msg[idx] * di * di + bias[c];
}

// agg[dst] += msg[src] * dinv[src]*dinv[dst]   (scatter over edges)
template <int NC>
__global__ void agg_edge(const float* __restrict__ msg, float* __restrict__ agg,
                         const int* __restrict__ src, const int* __restrict__ dst,
                         const float* __restrict__ dinv, long total) {
    long idx = blockIdx.x * (long)blockDim.x + threadIdx.x;
    if (idx >= total) return;
    long e = idx / NC;
    int  c = (int)(idx - e * NC);
    int s = src[e], d = dst[e];
    float nrm = dinv[s] * dinv[d];
    atomicAdd(&agg[(size_t)d * NC + c], msg[(size_t)s * NC + c] * nrm);
}

// BatchNorm statistics: per-channel sum & sumsq (LDS reduce then global atomics)
__global__ __launch_bounds__(256) void bn_stats(const float* __restrict__ x,
                                                float* __restrict__ stats) {
    __shared__ float ls[256];
    ls[threadIdx.x] = 0.f;
    __syncthreads();
    const long total = (long)NN * HH;
    for (long idx = blockIdx.x * (long)blockDim.x + threadIdx.x; idx < total;
         idx += (long)gridDim.x * blockDim.x) {
        float v = x[idx];
        int c = (int)(idx & (HH - 1));
        atomicAdd(&ls[c], v);
        atomicAdd(&ls[HH + c], v * v);
    }
    __syncthreads();
    atomicAdd(&stats[threadIdx.x], ls[threadIdx.x]);
}

__global__ void bn_finalize(const float* __restrict__ stats,
                            float* __restrict__ mean, float* __restrict__ rstd) {
    int c = threadIdx.x;
    float mu  = stats[c] * (1.f / NN);
    float var = stats[HH + c] * (1.f / NN) - mu * mu;
    mean[c] = mu;
    rstd[c] = rsqrtf(var + EPSB);
}

__global__ void bn_relu(const float* __restrict__ in, float* __restrict__ out,
                        const float* __restrict__ g, const float* __restrict__ be,
                        const float* __restrict__ mean, const float* __restrict__ rstd,
                        long total) {
    long idx = blockIdx.x * (long)blockDim.x + threadIdx.x;
    if (idx >= total) return;
    int c = (int)(idx & (HH - 1));
    float v = g[c] * (in[idx] - mean[c]) * rstd[c] + be[c];
    out[idx] = fmaxf(v, 0.f);
}

// log_softmax over 40 classes; one wave32 per row, lane j handles j and j+32
__global__ __launch_bounds__(256) void logsoftmax40(const float* __restrict__ in,
                                                    float* __restrict__ out) {
    int row  = blockIdx.x * (blockDim.x >> 5) + (threadIdx.x >> 5);
    int lane = threadIdx.x & 31;
    if (row >= NN) return;
    const float* p = in + (size_t)row * CC;
    float v0 = p[lane];
    bool  hi = (lane + 32) < CC;
    float v1 = hi ? p[lane + 32] : -3.4e38f;
    float m = fmaxf(v0, v1);
#pragma unroll
    for (int off = 16; off; off >>= 1) m = fmaxf(m, __shfl_xor(m, off, 32));
    float s = __expf(v0 - m) + (hi ? __expf(v1 - m) : 0.f);
#pragma unroll
    for (int off = 16; off; off >>= 1) s += __shfl_xor(s, off, 32);
    float lse = m + __logf(s);
    out[(size_t)row * CC + lane] = v0 - lse;
    if (hi) out[(size_t)row * CC + lane + 32] = v1 - lse;
}

// KENN clause layer over edges: softmax of 3 literals per (edge, class)
template <int NC>
__global__ void kenn_edge(const float* __restrict__ z, const float* __restrict__ w,
                          const int* __restrict__ src, const int* __restrict__ dst,
                          float* __restrict__ dz, float* __restrict__ bin_out,
                          const float* __restrict__ bin_in, long total) {
    long idx = blockIdx.x * (long)blockDim.x + threadIdx.x;
    if (idx >= total) return;
    long e = idx / NC;
    int  c = (int)(idx - e * NC);
    int s = src[e], d = dst[e];
    float zx = z[(size_t)s * NC + c];
    float zy = z[(size_t)d * NC + c];
    float b  = bin_in[e];
    float l0 = -zx, l1 = -b, l2 = zy;
    float m  = fmaxf(l0, fmaxf(l1, l2));
    float e0 = __expf(l0 - m), e1 = __expf(l1 - m), e2 = __expf(l2 - m);
    float scale = w[c] / (e0 + e1 + e2);
    atomicAdd(&dz[(size_t)s * NC + c], -e0 * scale);
    atomicAdd(&dz[(size_t)d * NC + c],  e2 * scale);
    atomicAdd(&bin_out[e], -e1 * scale);
}

__global__ void add_clear(const float* __restrict__ zin, float* __restrict__ dz,
                          float* __restrict__ zout, long n) {
    long i = blockIdx.x * (long)blockDim.x + threadIdx.x;
    if (i < n) { zout[i] = zin[i] + dz[i]; dz[i] = 0.f; }
}

// ---------------------------------------------------------------------------
// Host orchestration
// ---------------------------------------------------------------------------
static inline int gb(long n, int b) { return (int)((n + b - 1) / b); }

extern "C" void kernel_launch(void* const* d_in, const int* in_sizes, int n_in,
                              void* d_out, int out_size, void* d_ws, size_t ws_size,
                              hipStream_t stream) {
    (void)in_sizes; (void)n_in; (void)out_size; (void)ws_size;
    const float* x    = (const float*)d_in[0];
    const int*   src  = (const int*)d_in[1];
    const int*   dstp = src + EE;
    const float* rel  = (const float*)d_in[2];
    const float* W0   = (const float*)d_in[3];
    const float* b0   = (const float*)d_in[4];
    const float* W1   = (const float*)d_in[5];
    const float* b1   = (const float*)d_in[6];
    const float* W2   = (const float*)d_in[7];
    const float* b2   = (const float*)d_in[8];
    const float* g0   = (const float*)d_in[9];
    const float* be0  = (const float*)d_in[10];
    const float* g1   = (const float*)d_in[11];
    const float* be1  = (const float*)d_in[12];
    const float* cw   = (const float*)d_in[13];

    float* ws = (float*)d_ws;
    size_t off = 0;
#define WSA(name, cnt) float* name = ws + off; off += (((size_t)(cnt)) + 63) & ~(size_t)63;
    WSA(dinv,  NN)
    WSA(bufA,  (size_t)NN * HH)
    WSA(bufB,  (size_t)NN * HH)
    WSA(bufC,  (size_t)NN * HH)
    WSA(msg2,  (size_t)NN * CC)
    WSA(z2,    (size_t)NN * CC)
    WSA(zP,    (size_t)NN * CC)
    WSA(zQ,    (size_t)NN * CC)
    WSA(dz,    (size_t)NN * CC)
    WSA(bin0,  EE)
    WSA(bin1,  EE)
    WSA(stats, 2 * HH)
    WSA(meanb, HH)
    WSA(rstdb, HH)
#undef WSA

    const long NH  = (long)NN * HH, EH = (long)EE * HH;
    const long NC_ = (long)NN * CC, EC = (long)EE * CC;

    // degree / symmetric norm: deg = 1 + incoming; dinv = rsqrt(deg)
    fill_kernel <<<gb(NN, 256), 256, 0, stream>>>(dinv, 1.f, NN);
    deg_kernel  <<<gb(EE, 256), 256, 0, stream>>>(dinv, dstp);
    rsqrt_kernel<<<gb(NN, 256), 256, 0, stream>>>(dinv, NN);

    // ---- GCN conv 0 + BN + ReLU ----
    gemm_wmma<HH, 8><<<NN / MROWS, 256, 0, stream>>>(x, W0, bufA);
    agg_init<HH><<<gb(NH, 256), 256, 0, stream>>>(bufA, bufB, dinv, b0, NH);
    agg_edge<HH><<<gb(EH, 256), 256, 0, stream>>>(bufA, bufB, src, dstp, dinv, EH);
    fill_kernel <<<1, 256, 0, stream>>>(stats, 0.f, 2 * HH);
    bn_stats    <<<1024, 256, 0, stream>>>(bufB, stats);
    bn_finalize <<<1, HH, 0, stream>>>(stats, meanb, rstdb);
    bn_relu     <<<gb(NH, 256), 256, 0, stream>>>(bufB, bufA, g0, be0, meanb, rstdb, NH);

    // ---- GCN conv 1 + BN + ReLU ----
    gemm_wmma<HH, 8><<<NN / MROWS, 256, 0, stream>>>(bufA, W1, bufB);
    agg_init<HH><<<gb(NH, 256), 256, 0, stream>>>(bufB, bufC, dinv, b1, NH);
    agg_edge<HH><<<gb(EH, 256), 256, 0, stream>>>(bufB, bufC, src, dstp, dinv, EH);
    fill_kernel <<<1, 256, 0, stream>>>(stats, 0.f, 2 * HH);
    bn_stats    <<<1024, 256, 0, stream>>>(bufC, stats);
    bn_finalize <<<1, HH, 0, stream>>>(stats, meanb, rstdb);
    bn_relu     <<<gb(NH, 256), 256, 0, stream>>>(bufC, bufA, g1, be1, meanb, rstdb, NH);

    // ---- GCN conv 2 (H -> 40) + log_softmax ----
    gemm_wmma<CC, 3><<<NN / MROWS, 96, 0, stream>>>(bufA, W2, msg2);
    agg_init<CC><<<gb(NC_, 256), 256, 0, stream>>>(msg2, z2, dinv, b2, NC_);
    agg_edge<CC><<<gb(EC, 256), 256, 0, stream>>>(msg2, z2, src, dstp, dinv, EC);
    logsoftmax40<<<gb(NN, 8), 256, 0, stream>>>(z2, zP);

    // ---- KENN layers ----
    copy_kernel<<<gb(EE, 256), 256, 0, stream>>>(bin0, rel, EE);
    fill_kernel<<<gb(NC_, 256), 256, 0, stream>>>(dz, 0.f, NC_);
    float* zin = zP; float* zout = zQ; float* bi = bin0; float* bo = bin1;
    for (int l = 0; l < NUM_KENN; ++l) {
        copy_kernel<<<gb(EE, 256), 256, 0, stream>>>(bo, bi, EE);
        kenn_edge<CC><<<gb(EC, 256), 256, 0, stream>>>(zin, cw + l * CC, src, dstp,
                                                       dz, bo, bi, EC);
        add_clear<<<gb(NC_, 256), 256, 0, stream>>>(zin, dz, zout, NC_);
        float* t = zin; zin = zout; zout = t;
        t = bi; bi = bo; bo = t;
    }

    // final log_softmax straight into d_out (fp32)
    logsoftmax40<<<gb(NN, 8), 256, 0, stream>>>(zin, (float*)d_out);
}